// SEIR_RNN_33818572489192
// MI455X (gfx1250) — compile-verified
//
#include <hip/hip_runtime.h>
#include <stdint.h>

// SEIR RNN, BATCH=65536, STEPS=200, out (B,200,1,4) f32.
// Store-bandwidth bound: ~210 MB written -> ~9us floor at 23.3 TB/s.
// One thread = one trajectory (state in VGPRs). Each wave stages 8 steps x
// 32 elements in LDS (ds_store_b128), then flushes with CDNA5 async
// LDS->global B128 stores; lane remap makes every flush instruction write
// four full, aligned 128B lines. Double-buffered LDS with ASYNCcnt waits.

#define STEPS   200
#define CHUNK   8
#define NCHUNKS (STEPS / CHUNK)      // 25
#define BLOCK   256
#define WAVES   (BLOCK / 32)         // 8 waves (wave32)
#define TSTRIDE 33                   // float4 stride between step rows (bank pad)
#define WTILE   (CHUNK * TSTRIDE)    // 264 float4 per wave per buffer
#define BUFSZ   (WAVES * WTILE)      // float4 per buffer (all waves)

#define N_POP_F 100000.0f
#define CLAMP_F 100000.0f
#define H_F     0.5f

__global__ __launch_bounds__(BLOCK) void seir_rnn_kernel(
    const float* __restrict__ init,   // (BATCH,1,4)
    const float* __restrict__ w,      // (7,)
    float* __restrict__ out)          // (BATCH,STEPS,1,4)
{
    __shared__ float4 lds[2][WAVES][WTILE];

    const int lane = threadIdx.x & 31;
    const int wave = threadIdx.x >> 5;
    const int gid  = blockIdx.x * BLOCK + (int)threadIdx.x;   // trajectory id

    const float A = w[4];
    const float B = w[5];
    const float C = w[6];

    float4 x0 = ((const float4*)init)[gid];
    float s = x0.x, e = x0.y, i = x0.z, r = x0.w;

    // Flush lane remap: 8 consecutive lanes cover one element's 8 steps
    // (one aligned 128B line); each instruction covers 4 elements = 4 lines.
    const int ft = lane & 7;      // step within chunk this lane flushes
    const int fe = lane >> 3;     // element sub-index within a 4-element group
    const int elem_base = blockIdx.x * BLOCK + wave * 32;

    // Per-lane 64-bit global flush addresses for j=0..7 at chunk 0;
    // each advances by CHUNK*16 = 128 bytes per chunk.
    unsigned long long ga[8];
    #pragma unroll
    for (int j = 0; j < 8; ++j) {
        size_t q = (size_t)(unsigned)(elem_base + fe + 4 * j) * STEPS + (unsigned)ft;
        ga[j] = (unsigned long long)(uintptr_t)((const float4*)out + q);
    }

    // This lane's LDS flush-source byte offset (buf 0, j = 0).
    const unsigned lds_base = (unsigned)(uintptr_t)&lds[0][0][0];
    const unsigned lflush0 =
        lds_base + (unsigned)(wave * WTILE + ft * TSTRIDE + fe) * 16u;

    for (int c = 0; c < NCHUNKS; ++c) {
        const int buf = c & 1;
        // Buffer `buf` was flushed 2 chunks ago; allow only the most recent
        // chunk's 8 async stores to remain in flight before overwriting it.
        asm volatile("s_wait_asynccnt 0x8" ::: "memory");

        #pragma unroll
        for (int t = 0; t < CHUNK; ++t) {
            float infl = B * s * (e + i) / N_POP_F;
            float d0 = -infl;
            float d1 = infl - A * e;
            float d2 = A * e - C * i;
            float d3 = C * i;
            d0 = fminf(fmaxf(d0, -CLAMP_F), CLAMP_F);
            d1 = fminf(fmaxf(d1, -CLAMP_F), CLAMP_F);
            d2 = fminf(fmaxf(d2, -CLAMP_F), CLAMP_F);
            d3 = fminf(fmaxf(d3, -CLAMP_F), CLAMP_F);
            s += H_F * d0;
            e += H_F * d1;
            i += H_F * d2;
            r += H_F * d3;
            // Direct shared-array indexing -> ds_store_b128 (stays addrspace(3)).
            lds[buf][wave][t * TSTRIDE + lane] = make_float4(s, e, i, r);
        }

        // Our DS writes must land before the async engine reads LDS.
        asm volatile("s_wait_dscnt 0x0" ::: "memory");

        const unsigned lf = lflush0 + (unsigned)buf * (unsigned)(BUFSZ * 16);
        #pragma unroll
        for (int j = 0; j < 8; ++j) {
            unsigned ldsoff = lf + 64u * (unsigned)j;   // eloc = fe + 4j -> +64B per j
            asm volatile("global_store_async_from_lds_b128 %0, %1, off"
                         :: "v"(ga[j]), "v"(ldsoff)
                         : "memory");
            ga[j] += 128ull;                            // next chunk: +8 float4
        }
    }
    asm volatile("s_wait_asynccnt 0x0" ::: "memory");
}

extern "C" void kernel_launch(void* const* d_in, const int* in_sizes, int n_in,
                              void* d_out, int out_size, void* d_ws, size_t ws_size,
                              hipStream_t stream) {
    const float* init = (const float*)d_in[0];   // BATCH*1*4 floats
    const float* wts  = (const float*)d_in[1];   // 7 floats
    float* out = (float*)d_out;                  // BATCH*STEPS*1*4 floats

    const int n = in_sizes[0] / 4;               // BATCH = 65536
    const int grid = n / BLOCK;                  // 256 blocks of 256 threads
    seir_rnn_kernel<<<grid, BLOCK, 0, stream>>>(init, wts, out);
}